// GATBlock_34711925686354
// MI455X (gfx1250) — compile-verified
//
#include <hip/hip_runtime.h>
#include <cfloat>

typedef __attribute__((ext_vector_type(2))) float v2f;
typedef __attribute__((ext_vector_type(8))) float v8f;

#define HID   128
#define HEADS 8
#define HDIM  16

// ---------- helpers ----------
__device__ __forceinline__ unsigned enc_ord(float f) {
    unsigned u = __float_as_uint(f);
    return u ^ ((unsigned)((int)u >> 31) | 0x80000000u);
}
__device__ __forceinline__ float dec_ord(unsigned u) {
    unsigned v = (u & 0x80000000u) ? (u ^ 0x80000000u) : ~u;
    return __uint_as_float(v);
}
__device__ __forceinline__ float leaky(float x) { return x > 0.f ? x : 0.2f * x; }

// ---------- 0: init accumulators (must run every call: graph replays) ----------
__global__ __launch_bounds__(256) void init_kernel(float* deg, float* lsum, float* we,
                                                   float* denom, unsigned* m, float* agg,
                                                   const float* w_edge, const float* att_edge,
                                                   int N) {
    long long idx = (long long)blockIdx.x * blockDim.x + threadIdx.x;
    if (idx < (long long)N * HID) agg[idx] = 0.f;
    if (idx < (long long)N * HEADS) { denom[idx] = 0.f; m[idx] = 0u; }  // 0 < enc(any float)
    if (idx < N) { deg[idx] = 0.f; lsum[idx] = 0.f; }
    if (idx < HEADS) {
        float s = 0.f;
        #pragma unroll
        for (int c = 0; c < HDIM; ++c) s += w_edge[idx * HDIM + c] * att_edge[idx * HDIM + c];
        we[idx] = s;
    }
}

// ---------- 1: in-degree + loop-attr sums ----------
__global__ __launch_bounds__(256) void deg_kernel(const long long* ei, const float* ea,
                                                  float* deg, float* lsum, int E) {
    int e = blockIdx.x * blockDim.x + threadIdx.x;
    if (e >= E) return;
    int dst = (int)ei[(long long)E + e];
    atomicAdd(deg + dst, 1.f);
    atomicAdd(lsum + dst, ea[e]);
}

// ---------- 2: h = x @ W via V_WMMA_F32_16X16X4_F32 ----------
// block = 256 threads = 8 waves; block covers a 16-node row strip, wave w does cols [16w,16w+16)
__global__ __launch_bounds__(256) void gemm_kernel(const float* __restrict__ x,
                                                   const float* __restrict__ W,
                                                   float* __restrict__ h, int N) {
    __shared__ float sX[16 * 132];           // padded stride 132 to avoid bank conflicts
    const int tid  = threadIdx.x;
    const int wave = tid >> 5;
    const int lane = tid & 31;
    const int m0 = blockIdx.x * 16;

    for (int i = tid; i < 16 * HID; i += 256) {
        int r = i >> 7, c = i & 127;
        int row = m0 + r;
        sX[r * 132 + c] = (row < N) ? x[(long long)row * HID + c] : 0.f;
    }
    __syncthreads();

    const int n0    = wave * 16;
    const int mrow  = lane & 15;       // A: M index / B,C,D: N index
    const int khalf = (lane >> 4) * 2; // lanes 16-31 hold K+2,K+3

    v8f c = {};
    #pragma unroll
    for (int k0 = 0; k0 < HID; k0 += 4) {
        v2f a, b;
        a.x = sX[mrow * 132 + k0 + khalf];
        a.y = sX[mrow * 132 + k0 + khalf + 1];
        b.x = W[(long long)(k0 + khalf) * HID + n0 + mrow];       // W hot in L2 (64 KB)
        b.y = W[(long long)(k0 + khalf + 1) * HID + n0 + mrow];
        c = __builtin_amdgcn_wmma_f32_16x16x4_f32(false, a, false, b,
                                                  (short)0, c, false, false);
    }

    const int mhi = (lane >> 4) * 8;
    #pragma unroll
    for (int r = 0; r < 8; ++r) {
        int row = m0 + r + mhi;
        if (row < N) h[(long long)row * HID + n0 + mrow] = c[r];
    }
}

// ---------- 3: per-node attention logits ----------
__global__ __launch_bounds__(256) void node_att_kernel(const float* __restrict__ h,
                                                       const float* __restrict__ att_src,
                                                       const float* __restrict__ att_dst,
                                                       float* a_src, float* a_dst, int N) {
    int idx = blockIdx.x * blockDim.x + threadIdx.x;   // idx = n*8 + head
    if (idx >= N * HEADS) return;
    int head = idx & 7;
    const float* hp = h + (long long)(idx >> 3) * HID + head * HDIM;
    const float* as = att_src + head * HDIM;
    const float* ad = att_dst + head * HDIM;
    float s = 0.f, d = 0.f;
    #pragma unroll
    for (int c = 0; c < HDIM; ++c) { float v = hp[c]; s += v * as[c]; d += v * ad[c]; }
    a_src[idx] = s; a_dst[idx] = d;
}

// ---------- edge alpha helper ----------
__device__ __forceinline__ void edge_decode(const long long* ei, const float* ea,
                                            const float* deg, const float* lsum,
                                            int e, int E, int& s, int& d, float& attr) {
    if (e < E) { s = (int)ei[e]; d = (int)ei[(long long)E + e]; attr = ea[e]; }
    else       { s = d = e - E; attr = lsum[s] / fmaxf(deg[s], 1.f); }
}

// ---------- 4: segment max ----------
__global__ __launch_bounds__(256) void edge_max_kernel(const long long* ei, const float* ea,
                                                       const float* deg, const float* lsum,
                                                       const float* we, const float* a_src,
                                                       const float* a_dst, unsigned* m,
                                                       int E, int N) {
    int e = blockIdx.x * blockDim.x + threadIdx.x;
    if (e >= E + N) return;
    int s, d; float attr;
    edge_decode(ei, ea, deg, lsum, e, E, s, d, attr);
    const float* as = a_src + (long long)s * HEADS;
    const float* ad = a_dst + (long long)d * HEADS;
    unsigned* mp = m + (long long)d * HEADS;
    #pragma unroll
    for (int hh = 0; hh < HEADS; ++hh) {
        float al = leaky(as[hh] + ad[hh] + attr * we[hh]);
        atomicMax(mp + hh, enc_ord(al));
    }
}

// ---------- 5: segment exp-sum ----------
__global__ __launch_bounds__(256) void edge_expsum_kernel(const long long* ei, const float* ea,
                                                          const float* deg, const float* lsum,
                                                          const float* we, const float* a_src,
                                                          const float* a_dst, const unsigned* m,
                                                          float* denom, int E, int N) {
    int e = blockIdx.x * blockDim.x + threadIdx.x;
    if (e >= E + N) return;
    int s, d; float attr;
    edge_decode(ei, ea, deg, lsum, e, E, s, d, attr);
    const float* as = a_src + (long long)s * HEADS;
    const float* ad = a_dst + (long long)d * HEADS;
    const unsigned* mp = m + (long long)d * HEADS;
    float* dn = denom + (long long)d * HEADS;
    #pragma unroll
    for (int hh = 0; hh < HEADS; ++hh) {
        float al = leaky(as[hh] + ad[hh] + attr * we[hh]);
        atomicAdd(dn + hh, expf(al - dec_ord(mp[hh])));
    }
}

// ---------- 6: weighted scatter-aggregate, one wave32 per edge ----------
__global__ __launch_bounds__(256) void edge_agg_kernel(const long long* ei, const float* ea,
                                                       const float* deg, const float* lsum,
                                                       const float* we, const float* a_src,
                                                       const float* a_dst, const unsigned* m,
                                                       const float* denom, const float* __restrict__ h,
                                                       float* agg, int E, int N) {
    long long gtid = (long long)blockIdx.x * blockDim.x + threadIdx.x;
    int e = (int)(gtid >> 5);
    int lane = threadIdx.x & 31;
    if (e >= E + N) return;
    int s, d; float attr;
    edge_decode(ei, ea, deg, lsum, e, E, s, d, attr);
    int hh = lane >> 2;                         // 4 lanes (=4 channels each) per head
    float al = leaky(a_src[(long long)s * HEADS + hh] +
                     a_dst[(long long)d * HEADS + hh] + attr * we[hh]);
    float mm = dec_ord(m[(long long)d * HEADS + hh]);
    float p  = expf(al - mm) / (denom[(long long)d * HEADS + hh] + 1e-16f);
    const float4 hv = *(const float4*)(h + (long long)s * HID + lane * 4);  // L2-resident gather
    float* ap = agg + (long long)d * HID + lane * 4;
    atomicAdd(ap + 0, p * hv.x);
    atomicAdd(ap + 1, p * hv.y);
    atomicAdd(ap + 2, p * hv.z);
    atomicAdd(ap + 3, p * hv.w);
}

// ---------- 7: bias + residual + LayerNorm + ReLU, one wave32 per node ----------
__global__ __launch_bounds__(256) void final_kernel(const float* __restrict__ agg,
                                                    const float* __restrict__ x,
                                                    const float* bias, const float* gamma,
                                                    const float* beta, float* out, int N) {
    long long gtid = (long long)blockIdx.x * blockDim.x + threadIdx.x;
    int n = (int)(gtid >> 5);
    int lane = threadIdx.x & 31;
    if (n >= N) return;
    const float4 av = *(const float4*)(agg + (long long)n * HID + lane * 4);
    const float4 xv = *(const float4*)(x   + (long long)n * HID + lane * 4);
    const float4 bv = *(const float4*)(bias + lane * 4);
    float y0 = av.x + bv.x + xv.x, y1 = av.y + bv.y + xv.y;
    float y2 = av.z + bv.z + xv.z, y3 = av.w + bv.w + xv.w;
    float sum = y0 + y1 + y2 + y3;
    #pragma unroll
    for (int o = 16; o >= 1; o >>= 1) sum += __shfl_xor(sum, o, 32);
    float mu = sum * (1.f / HID);
    float d0 = y0 - mu, d1 = y1 - mu, d2 = y2 - mu, d3 = y3 - mu;
    float vs = d0 * d0 + d1 * d1 + d2 * d2 + d3 * d3;
    #pragma unroll
    for (int o = 16; o >= 1; o >>= 1) vs += __shfl_xor(vs, o, 32);
    float inv = rsqrtf(vs * (1.f / HID) + 1e-5f);
    const float4 gv = *(const float4*)(gamma + lane * 4);
    const float4 be = *(const float4*)(beta + lane * 4);
    float4 r;
    r.x = fmaxf(d0 * inv * gv.x + be.x, 0.f);
    r.y = fmaxf(d1 * inv * gv.y + be.y, 0.f);
    r.z = fmaxf(d2 * inv * gv.z + be.z, 0.f);
    r.w = fmaxf(d3 * inv * gv.w + be.w, 0.f);
    *(float4*)(out + (long long)n * HID + lane * 4) = r;
}

// ---------- host ----------
extern "C" void kernel_launch(void* const* d_in, const int* in_sizes, int n_in,
                              void* d_out, int out_size, void* d_ws, size_t ws_size,
                              hipStream_t stream) {
    const float*     x        = (const float*)d_in[0];
    const long long* ei       = (const long long*)d_in[1];
    const float*     ea       = (const float*)d_in[2];
    const float*     W        = (const float*)d_in[3];
    const float*     att_src  = (const float*)d_in[4];
    const float*     att_dst  = (const float*)d_in[5];
    const float*     w_edge   = (const float*)d_in[6];
    const float*     att_edge = (const float*)d_in[7];
    const float*     bias     = (const float*)d_in[8];
    const float*     gamma    = (const float*)d_in[9];
    const float*     beta     = (const float*)d_in[10];

    const int N = in_sizes[0] / HID;
    const int E = in_sizes[2];

    float* ws = (float*)d_ws;
    size_t off = 0;
    float*    deg   = ws + off; off += (size_t)N;
    float*    lsum  = ws + off; off += (size_t)N;
    float*    we    = ws + off; off += 16;
    float*    a_src = ws + off; off += (size_t)N * HEADS;
    float*    a_dst = ws + off; off += (size_t)N * HEADS;
    unsigned* m     = (unsigned*)(ws + off); off += (size_t)N * HEADS;
    float*    denom = ws + off; off += (size_t)N * HEADS;
    float*    h     = ws + off; off += (size_t)N * HID;
    float*    agg   = ws + off; off += (size_t)N * HID;
    float*    out   = (float*)d_out;

    const int T = 256;
    long long initN = (long long)N * HID;
    init_kernel<<<(int)((initN + T - 1) / T), T, 0, stream>>>(deg, lsum, we, denom, m, agg,
                                                              w_edge, att_edge, N);
    deg_kernel<<<(E + T - 1) / T, T, 0, stream>>>(ei, ea, deg, lsum, E);
    gemm_kernel<<<(N + 15) / 16, T, 0, stream>>>(x, W, h, N);
    node_att_kernel<<<(N * HEADS + T - 1) / T, T, 0, stream>>>(h, att_src, att_dst, a_src, a_dst, N);
    edge_max_kernel<<<(E + N + T - 1) / T, T, 0, stream>>>(ei, ea, deg, lsum, we,
                                                           a_src, a_dst, m, E, N);
    edge_expsum_kernel<<<(E + N + T - 1) / T, T, 0, stream>>>(ei, ea, deg, lsum, we,
                                                              a_src, a_dst, m, denom, E, N);
    long long aggT = (long long)(E + N) * 32;
    edge_agg_kernel<<<(int)((aggT + T - 1) / T), T, 0, stream>>>(ei, ea, deg, lsum, we,
                                                                 a_src, a_dst, m, denom,
                                                                 h, agg, E, N);
    long long finT = (long long)N * 32;
    final_kernel<<<(int)((finT + T - 1) / T), T, 0, stream>>>(agg, x, bias, gamma, beta, out, N);
}